// Conv2d_Custom_59167469470407
// MI455X (gfx1250) — compile-verified
//
#include <hip/hip_runtime.h>

typedef float v2f __attribute__((ext_vector_type(2)));
typedef float v8f __attribute__((ext_vector_type(8)));

#define B_N    8
#define C_IN   16
#define C_OUT  16
#define H_IN   512
#define W_IN   512
#define KS     3
#define H_OUT  510
#define W_OUT  510
#define L_OUT  (H_OUT * W_OUT)          // 260100
#define TILES_X   32                    // ceil(510/16)
#define TILES_ALL (B_N * H_OUT * TILES_X)  // 130560

__global__ __launch_bounds__(256)
void conv3x3_wmma_f32(const float* __restrict__ x,
                      const float* __restrict__ w,
                      float* __restrict__ out) {
  const int lane   = threadIdx.x & 31;
  const int laneHi = lane >> 4;          // 0: lanes 0-15, 1: lanes 16-31
  const int n      = lane & 15;          // column-in-tile / weight row (M)

  const int waveInBlk = threadIdx.x >> 5;
  const int gwave  = blockIdx.x * (blockDim.x >> 5) + waveInBlk;
  const int nwaves = gridDim.x * (blockDim.x >> 5);

  // ---- Preload entire A matrix (weights, 16x144 f32) as 36 K=4 fragments.
  // K ordering: k = (kh*3+kw)*16 + ci.  A-frag layout per ISA (16x4 f32):
  //   VGPR0: lanes 0-15 -> K=k0,   lanes 16-31 -> K=k0+2
  //   VGPR1: lanes 0-15 -> K=k0+1, lanes 16-31 -> K=k0+3
  v2f afrag[36];
  {
    const int m = n;                      // output channel row of A
    #pragma unroll
    for (int kh = 0; kh < 3; ++kh) {
      #pragma unroll
      for (int kw = 0; kw < 3; ++kw) {
        #pragma unroll
        for (int c = 0; c < 4; ++c) {
          const int q   = (kh * 3 + kw) * 4 + c;
          const int ci0 = 4 * c + (laneHi ? 2 : 0);
          const float* wp = w + m * (C_IN * 9) + ci0 * 9 + kh * 3 + kw;
          v2f a;
          a.x = wp[0];                    // ci0
          a.y = wp[9];                    // ci0 + 1 (stride 9 in weight layout)
          afrag[q] = a;
        }
      }
    }
  }

  const int ciStride = H_IN * W_IN;       // channel stride in x (floats)

  // ---- Grid-stride loop over 16x16 output tiles (one tile per wave).
  for (int tile = gwave; tile < TILES_ALL; tile += nwaves) {
    const int tx   = tile % TILES_X;
    const int rest = tile / TILES_X;
    const int oh   = rest % H_OUT;
    const int b    = rest / H_OUT;

    const int owbase = tx * 16;
    const int col    = owbase + n;                       // desired output col
    const int colr   = (col <= (W_IN - KS)) ? col : (W_IN - KS); // clamp reads

    const float* xb = x + (size_t)b * C_IN * ciStride
                        + (size_t)laneHi * 2 * ciStride  // +2 channels for hi half
                        + colr;

    v8f acc = {0.f, 0.f, 0.f, 0.f, 0.f, 0.f, 0.f, 0.f};

    #pragma unroll
    for (int kh = 0; kh < 3; ++kh) {
      const float* xr = xb + (size_t)(oh + kh) * W_IN;
      #pragma unroll
      for (int kw = 0; kw < 3; ++kw) {
        #pragma unroll
        for (int c = 0; c < 4; ++c) {
          // B-frag (4x16 f32): VGPR0 = rows k0 / k0+2, VGPR1 = rows k0+1 / k0+3
          // row k0+j  <=>  input channel ci0+j, ci0 = 4c + 2*laneHi
          const float* p = xr + kw + (size_t)(4 * c) * ciStride;
          v2f bfrag;
          bfrag.x = p[0];
          bfrag.y = p[ciStride];
          acc = __builtin_amdgcn_wmma_f32_16x16x4_f32(
                    /*neg_a=*/false, afrag[(kh * 3 + kw) * 4 + c],
                    /*neg_b=*/false, bfrag,
                    /*c_mod=*/(short)0, acc,
                    /*reuse_a=*/false, /*reuse_b=*/false);
        }
      }
    }

    // ---- Store D (16x16 f32): VGPR r -> co=r (lanes 0-15), co=r+8 (lanes 16-31)
    // Output is write-once / never re-read: use non-temporal stores so the
    // 133 MB of output doesn't evict the (L2-resident, ~9x-reused) 134 MB input.
    if (col < W_OUT) {
      float* ob = out + (size_t)b * C_OUT * L_OUT
                      + (size_t)oh * W_OUT + owbase + n
                      + (size_t)(laneHi * 8) * L_OUT;
      #pragma unroll
      for (int r = 0; r < 8; ++r) {
        __builtin_nontemporal_store(acc[r], &ob[(size_t)r * L_OUT]);
      }
    }
  }
}

extern "C" void kernel_launch(void* const* d_in, const int* in_sizes, int n_in,
                              void* d_out, int out_size, void* d_ws, size_t ws_size,
                              hipStream_t stream) {
  const float* x = (const float*)d_in[0];   // (8, 16, 512, 512) f32
  const float* w = (const float*)d_in[1];   // (16, 16, 3, 3)    f32
  float* out = (float*)d_out;               // (8, 16, 260100)   f32

  dim3 block(256);                          // 8 wave32 waves
  dim3 grid(2048);                          // 16384 waves, ~8 tiles each
  conv3x3_wmma_f32<<<grid, block, 0, stream>>>(x, w, out);
}